// RWKVDecoder_30408368455717
// MI455X (gfx1250) — compile-verified
//
#include <hip/hip_runtime.h>

// ---------------------------------------------------------------------------
// RWKV single-token decode for MI455X (gfx1250, wave32).
// Memory-bound (~1.5 GB of f32 weights read once per token; ~65us roofline at
// 23.3 TB/s). Strategy:
//   * All matvecs via V_WMMA_F32_16X16X4_F32: A = 16x4 weight tile (each lane
//     loads one b64 = exactly its share of the tile, so every weight byte is
//     read once), B = input vector broadcast across the 16 columns. D columns
//     are identical; lanes 0/16 each hold 8 contiguous output rows.
//   * One persistent kernel; the 24-layer recurrence (~170 dependent phases)
//     is sequenced with a device-wide counter barrier (agent-scope fences +
//     s_sleep spin) instead of ~170 graph node dispatches.
// ---------------------------------------------------------------------------

#define Lc 24
#define Dc 1024
#define Hc 4096
#define Vc 50257

#define NBLOCKS  96
#define NTHREADS 256
#define WPB      (NTHREADS / 32)
#define NWAVES   (NBLOCKS * WPB)

typedef __attribute__((ext_vector_type(2))) float v2f;
typedef __attribute__((ext_vector_type(8))) float v8f;

struct Params {
  const int*   token;
  const float* state;
  const float* emb;
  const float* ln0_w; const float* ln0_b;
  const float* ln1_w; const float* ln1_b;
  const float* ln2_w; const float* ln2_b;
  const float* mix_k; const float* mix_v; const float* mix_r;
  const float* time_first; const float* time_decay;
  const float* att_kw; const float* att_vw; const float* att_rw; const float* att_ow;
  const float* ffn_mix_k; const float* ffn_mix_r;
  const float* ffn_kw; const float* ffn_vw; const float* ffn_rw;
  const float* lnout_w; const float* lnout_b;
  const float* head;
  float* logits;      // d_out[0..V)
  float* new_state;   // d_out[V..V+L*5*D)
  float* ws;          // float workspace (after barrier region)
  unsigned* barrier;  // grid barrier counter (zeroed each call)
};

__device__ __forceinline__ float sigm(float x) { return 1.0f / (1.0f + __expf(-x)); }

// ---- device-wide barrier -------------------------------------------------
__device__ __forceinline__ void grid_sync(unsigned* cnt, unsigned& tgt) {
  tgt += NBLOCKS;
  __syncthreads();
  if (threadIdx.x == 0) {
    __builtin_amdgcn_fence(__ATOMIC_RELEASE, "agent");
    __hip_atomic_fetch_add(cnt, 1u, __ATOMIC_RELAXED, __HIP_MEMORY_SCOPE_AGENT);
    while (__hip_atomic_load(cnt, __ATOMIC_RELAXED, __HIP_MEMORY_SCOPE_AGENT) < tgt)
      __builtin_amdgcn_s_sleep(1);
  }
  __syncthreads();
  __builtin_amdgcn_fence(__ATOMIC_ACQUIRE, "agent");  // invalidate stale L0 lines
}

// ---- WMMA matvec: one wave computes 16 output rows -----------------------
// A (16x4, M x K): lane m=lane&15 holds W[row0+m][k+koff .. k+koff+1],
//                  koff = (lane>>4)*2   -> one global b64 load per lane/WMMA.
// B (4x16, K x N): all columns = x, so vgpr0 = x[k+koff], vgpr1 = x[k+koff+1].
// D (16x16): vgpr j -> (M=j, lanes 0-15) / (M=8+j, lanes 16-31); columns equal.
__device__ __forceinline__ v8f mv_tile16(const float* __restrict__ W,
                                         const float* __restrict__ x,
                                         int N, int M, int row0, int lane) {
  int m = row0 + (lane & 15);
  m = (m < M) ? m : (M - 1);               // clamp for ragged last tile (head)
  const int koff = (lane >> 4) << 1;
  const float* wp = W + (size_t)m * N + koff;
  const float* xp = x + koff;
  v8f a0 = {0.f, 0.f, 0.f, 0.f, 0.f, 0.f, 0.f, 0.f};
  v8f a1 = a0;                              // two chains to hide WMMA latency
  for (int k = 0; k < N; k += 8) {
    v2f wa = *(const v2f*)(wp + k);
    v2f xa = *(const v2f*)(xp + k);
    v2f wb = *(const v2f*)(wp + k + 4);
    v2f xb = *(const v2f*)(xp + k + 4);
    a0 = __builtin_amdgcn_wmma_f32_16x16x4_f32(false, wa, false, xa, (short)0, a0, false, false);
    a1 = __builtin_amdgcn_wmma_f32_16x16x4_f32(false, wb, false, xb, (short)0, a1, false, false);
  }
  return a0 + a1;
}

// modes: 0 plain, 1 sigmoid, 2 relu^2, 3 y+=v, 4 y+=aux[r]*v
__device__ __forceinline__ void mv_epilogue(float* __restrict__ y, int M, int row0,
                                            int lane, v8f acc, int mode,
                                            const float* __restrict__ aux) {
  if ((lane & 15) == 0) {                   // lanes 0 and 16 own the results
    int base = row0 + ((lane >> 4) << 3);
#pragma unroll
    for (int j = 0; j < 8; ++j) {
      int rr = base + j;
      if (rr < M) {
        float v = acc[j];
        if (mode == 1)      v = sigm(v);
        else if (mode == 2) { float t = fmaxf(v, 0.f); v = t * t; }
        else if (mode == 3) v = y[rr] + v;
        else if (mode == 4) v = y[rr] + aux[rr] * v;
        y[rr] = v;
      }
    }
  }
}

__device__ __forceinline__ void mv_run(const float* __restrict__ W,
                                       const float* __restrict__ x,
                                       float* __restrict__ y,
                                       int M, int N, int mode,
                                       const float* __restrict__ aux,
                                       int wg, int lane) {
  int ntiles = (M + 15) >> 4;
  for (int t = wg; t < ntiles; t += NWAVES) {
    v8f acc = mv_tile16(W, x, N, M, t << 4, lane);
    mv_epilogue(y, M, t << 4, lane, acc, mode, aux);
  }
}

// ---- block-0 LayerNorm helpers (D=1024, 256 threads x 4 elems) ------------
__device__ __forceinline__ float block_sum(float v, float* s_red, int tid) {
#pragma unroll
  for (int off = 16; off; off >>= 1) v += __shfl_xor(v, off, 32);
  int wid = tid >> 5;
  if ((tid & 31) == 0) s_red[wid] = v;
  __syncthreads();
  if (tid == 0) {
    float s = 0.f;
    for (int i = 0; i < WPB; ++i) s += s_red[i];
    s_red[0] = s;
  }
  __syncthreads();
  float r = s_red[0];
  __syncthreads();
  return r;
}

__device__ __forceinline__ void block_layernorm(const float* __restrict__ in,
                                                float* s_red, int tid,
                                                float vals[4], float& mu, float& rstd) {
  float s = 0.f;
#pragma unroll
  for (int j = 0; j < 4; ++j) { vals[j] = in[tid + j * NTHREADS]; s += vals[j]; }
  mu = block_sum(s, s_red, tid) * (1.0f / Dc);
  float vs = 0.f;
#pragma unroll
  for (int j = 0; j < 4; ++j) { float d = vals[j] - mu; vs += d * d; }
  float var = block_sum(vs, s_red, tid) * (1.0f / Dc);
  rstd = rsqrtf(var + 1e-5f);
}

// ---------------------------------------------------------------------------
__global__ void __launch_bounds__(NTHREADS)
rwkv_persistent(Params P) {
  __shared__ float s_red[WPB];
  const int tid  = threadIdx.x;
  const int lane = tid & 31;
  const int wg   = blockIdx.x * WPB + (tid >> 5);
  unsigned tgt = 0;

  float* F    = P.ws;
  float* X    = F;                 // current residual x           [D]
  float* XN   = F + 1 * Dc;        // ln1(x)                       [D]
  float* INK  = F + 2 * Dc;        // mixed inputs                 [D] x3
  float* INV  = F + 3 * Dc;
  float* INR  = F + 4 * Dc;
  float* Kv   = F + 5 * Dc;        // k, v, r                      [D] x3
  float* Vv   = F + 6 * Dc;
  float* Rv   = F + 7 * Dc;
  float* RWKV = F + 8 * Dc;        // r * wkv                      [D]
  float* INKC = F + 9 * Dc;        // ffn mixed inputs             [D] x2
  float* INRC = F + 10 * Dc;
  float* RC   = F + 11 * Dc;       // sigmoid(frw@inrc)            [D]
  float* KC   = F + 12 * Dc;       // relu(fkw@inkc)^2             [H]
  float* XF   = F + 12 * Dc + Hc;  // ln_out(x)                    [D]

  // ---- phase 0: x = ln(emb[token], ln0) ----
  if (blockIdx.x == 0) {
    const float* row = P.emb + (size_t)(*P.token) * Dc;
    float vals[4], mu, rstd;
    block_layernorm(row, s_red, tid, vals, mu, rstd);
#pragma unroll
    for (int j = 0; j < 4; ++j) {
      int i = tid + j * NTHREADS;
      X[i] = (vals[j] - mu) * rstd * P.ln0_w[i] + P.ln0_b[i];
    }
  }
  grid_sync(P.barrier, tgt);

  const size_t DD = (size_t)Dc * Dc;
  for (int l = 0; l < Lc; ++l) {
    const float* kw  = P.att_kw + l * DD;
    const float* vw  = P.att_vw + l * DD;
    const float* rw  = P.att_rw + l * DD;
    const float* ow  = P.att_ow + l * DD;
    const float* fkw = P.ffn_kw + (size_t)l * Hc * Dc;
    const float* fvw = P.ffn_vw + (size_t)l * Dc * Hc;
    const float* frw = P.ffn_rw + l * DD;
    const float* os  = P.state     + (size_t)l * 5 * Dc;  // old state (read-only)
    float*       ns  = P.new_state + (size_t)l * 5 * Dc;  // new state (write-only)

    // ---- A: xn = ln1(x); time-mix inputs; new_s1 = xn ----
    if (blockIdx.x == 0) {
      float vals[4], mu, rstd;
      block_layernorm(X, s_red, tid, vals, mu, rstd);
      const float* w  = P.ln1_w + l * Dc;  const float* b  = P.ln1_b + l * Dc;
      const float* mk = P.mix_k + l * Dc;  const float* mv = P.mix_v + l * Dc;
      const float* mr = P.mix_r + l * Dc;
#pragma unroll
      for (int j = 0; j < 4; ++j) {
        int i = tid + j * NTHREADS;
        float xn = (vals[j] - mu) * rstd * w[i] + b[i];
        XN[i] = xn;  ns[Dc + i] = xn;
        float prev = os[Dc + i];
        INK[i] = xn * mk[i] + prev * (1.f - mk[i]);
        INV[i] = xn * mv[i] + prev * (1.f - mv[i]);
        INR[i] = xn * mr[i] + prev * (1.f - mr[i]);
      }
    }
    grid_sync(P.barrier, tgt);

    // ---- B: k = kw@ink, v = vw@inv, r = sigmoid(rw@inr) (192 tiles) ----
    for (int t = wg; t < 192; t += NWAVES) {
      int which = t >> 6, tt = (t & 63) << 4;
      const float* W  = (which == 0) ? kw  : (which == 1) ? vw  : rw;
      const float* xi = (which == 0) ? INK : (which == 1) ? INV : INR;
      float*       y  = (which == 0) ? Kv  : (which == 1) ? Vv  : Rv;
      v8f acc = mv_tile16(W, xi, Dc, Dc, tt, lane);
      mv_epilogue(y, Dc, tt, lane, acc, (which == 2) ? 1 : 0, nullptr);
    }
    grid_sync(P.barrier, tgt);

    // ---- C: WKV recurrence (elementwise over D) ----
    {
      int gi = blockIdx.x * NTHREADS + tid;
      if (gi < Dc) {
        const float* tf = P.time_first + l * Dc;
        const float* td = P.time_decay + l * Dc;
        float k_ = Kv[gi], v_ = Vv[gi], r_ = Rv[gi];
        float aa = os[2 * Dc + gi], bb = os[3 * Dc + gi], pp = os[4 * Dc + gi];
        float ww  = tf[gi] + k_;
        float qq  = fmaxf(pp, ww);
        float e1  = __expf(pp - qq), e2 = __expf(ww - qq);
        float wkv = (e1 * aa + e2 * v_) / (e1 * bb + e2);
        RWKV[gi] = r_ * wkv;
        float ww2 = pp + td[gi];
        float qq2 = fmaxf(ww2, k_);
        float f1  = __expf(ww2 - qq2), f2 = __expf(k_ - qq2);
        ns[2 * Dc + gi] = f1 * aa + f2 * v_;
        ns[3 * Dc + gi] = f1 * bb + f2;
        ns[4 * Dc + gi] = qq2;
      }
    }
    grid_sync(P.barrier, tgt);

    // ---- D: x += ow @ (r*wkv) ----
    mv_run(ow, RWKV, X, Dc, Dc, 3, nullptr, wg, lane);
    grid_sync(P.barrier, tgt);

    // ---- E: xn2 = ln2(x); ffn-mix inputs; new_s0 = xn2 ----
    if (blockIdx.x == 0) {
      float vals[4], mu, rstd;
      block_layernorm(X, s_red, tid, vals, mu, rstd);
      const float* w   = P.ln2_w + l * Dc;      const float* b   = P.ln2_b + l * Dc;
      const float* fmk = P.ffn_mix_k + l * Dc;  const float* fmr = P.ffn_mix_r + l * Dc;
#pragma unroll
      for (int j = 0; j < 4; ++j) {
        int i = tid + j * NTHREADS;
        float xn2 = (vals[j] - mu) * rstd * w[i] + b[i];
        ns[i] = xn2;
        float prev0 = os[i];
        INKC[i] = xn2 * fmk[i] + prev0 * (1.f - fmk[i]);
        INRC[i] = xn2 * fmr[i] + prev0 * (1.f - fmr[i]);
      }
    }
    grid_sync(P.barrier, tgt);

    // ---- F: kc = relu(fkw@inkc)^2 (256 tiles), rc = sigmoid(frw@inrc) (64) ----
    for (int t = wg; t < 320; t += NWAVES) {
      if (t < 256) {
        v8f acc = mv_tile16(fkw, INKC, Dc, Hc, t << 4, lane);
        mv_epilogue(KC, Hc, t << 4, lane, acc, 2, nullptr);
      } else {
        int tt = (t - 256) << 4;
        v8f acc = mv_tile16(frw, INRC, Dc, Dc, tt, lane);
        mv_epilogue(RC, Dc, tt, lane, acc, 1, nullptr);
      }
    }
    grid_sync(P.barrier, tgt);

    // ---- G: x += rc * (fvw @ kc)   (N = H = 4096) ----
    mv_run(fvw, KC, X, Dc, Hc, 4, RC, wg, lane);
    grid_sync(P.barrier, tgt);
  }

  // ---- final LN ----
  if (blockIdx.x == 0) {
    float vals[4], mu, rstd;
    block_layernorm(X, s_red, tid, vals, mu, rstd);
#pragma unroll
    for (int j = 0; j < 4; ++j) {
      int i = tid + j * NTHREADS;
      XF[i] = (vals[j] - mu) * rstd * P.lnout_w[i] + P.lnout_b[i];
    }
  }
  grid_sync(P.barrier, tgt);

  // ---- head: logits = head @ xf  (3142 tiles across 768 waves) ----
  mv_run(P.head, XF, P.logits, Vc, Dc, 0, nullptr, wg, lane);
}

__global__ void init_counter(unsigned* c) {
  if (threadIdx.x == 0) *c = 0u;
}

// ---------------------------------------------------------------------------
extern "C" void kernel_launch(void* const* d_in, const int* in_sizes, int n_in,
                              void* d_out, int out_size, void* d_ws, size_t ws_size,
                              hipStream_t stream) {
  Params P;
  P.token      = (const int*)  d_in[0];
  P.state      = (const float*)d_in[1];
  P.emb        = (const float*)d_in[2];
  P.ln0_w      = (const float*)d_in[3];  P.ln0_b = (const float*)d_in[4];
  P.ln1_w      = (const float*)d_in[5];  P.ln1_b = (const float*)d_in[6];
  P.ln2_w      = (const float*)d_in[7];  P.ln2_b = (const float*)d_in[8];
  P.mix_k      = (const float*)d_in[9];
  P.mix_v      = (const float*)d_in[10];
  P.mix_r      = (const float*)d_in[11];
  P.time_first = (const float*)d_in[12];
  P.time_decay = (const float*)d_in[13];
  P.att_kw     = (const float*)d_in[14];
  P.att_vw     = (const float*)d_in[15];
  P.att_rw     = (const float*)d_in[16];
  P.att_ow     = (const float*)d_in[17];
  P.ffn_mix_k  = (const float*)d_in[18];
  P.ffn_mix_r  = (const float*)d_in[19];
  P.ffn_kw     = (const float*)d_in[20];
  P.ffn_vw     = (const float*)d_in[21];
  P.ffn_rw     = (const float*)d_in[22];
  P.lnout_w    = (const float*)d_in[23]; P.lnout_b = (const float*)d_in[24];
  P.head       = (const float*)d_in[25];
  P.logits     = (float*)d_out;
  P.new_state  = (float*)d_out + Vc;
  P.barrier    = (unsigned*)d_ws;
  P.ws         = (float*)d_ws + 64;      // workspace floats after 256B barrier region

  init_counter<<<1, 32, 0, stream>>>(P.barrier);
  rwkv_persistent<<<NBLOCKS, NTHREADS, 0, stream>>>(P);
}